// TensorProduct_45268955300486
// MI455X (gfx1250) — compile-verified
//
#include <hip/hip_runtime.h>

// CDNA5 / gfx1250: exact-f32 CG tensor product via V_WMMA_F32_16X16X4_F32.
// out[n, m3, c] = sum_k cg[k] * x[n, mu1[k], c] * y[n, mu2[k], c]
// Dense matmul form: out[M, col] = sum_K W[K, M] * z[K, col],
//   K = i1*16+i2 (256), M = output row (99 -> 112 padded), col = (n, c).
// W is densified into LDS, K-pair interleaved so A fragments load as b64.

typedef __attribute__((ext_vector_type(2))) float v2f;
typedef __attribute__((ext_vector_type(8))) float v8f;

#define N_EDGES  4096
#define DIM_IN   16
#define CHANNELS 128
#define DIM_OUT  99
#define MPAD     112                 // 7 M-tiles of 16
#define MT       7
#define KDIM     256                 // 16*16 (i1,i2) pairs
#define KT       (KDIM / 4)          // 64 K-steps of 4 for f32 WMMA
#define WAVES    8
#define BLOCK    (WAVES * 32)
#define COLTILES (N_EDGES * CHANNELS / 16)   // 32768 column tiles of 16 (n,c) cols
#define GRID     1024
#define TS       20                  // tile row stride (16B aligned, bank-conflict free)

static_assert(COLTILES % (GRID * WAVES) == 0, "uniform tile loop required");

__global__ __launch_bounds__(BLOCK)
void tp_cg_wmma_kernel(const float* __restrict__ x, const float* __restrict__ y,
                       const int* __restrict__ mu1, const int* __restrict__ mu2,
                       const int* __restrict__ mu3, const float* __restrict__ cg,
                       int nnz, float* __restrict__ out) {
  // Dense CG matrix, K-pair interleaved: element (k, m) at Wp[k>>1][2m + (k&1)].
  // A fragment (K=kb, K=kb+1) for row m is one aligned b64 at &Wp[kb>>1][2m].
  __shared__ float Wp[KDIM / 2][MPAD * 2];                 // 112 KB
  __shared__ __align__(16) float xt [WAVES][DIM_IN][TS];   // x tile, i-major
  __shared__ __align__(16) float ytp[WAVES][16][TS];       // y tile, transposed (j-major)

  const int tid  = threadIdx.x;
  const int wave = tid >> 5;
  const int lane = tid & 31;
  const int j    = lane & 15;   // N column within tile (and M row within A tile)
  const int hi   = lane >> 4;   // lane-half selects K offset {0,2}

  // ---- build dense padded CG matrix in LDS from sparse lists ----
  for (int i = tid; i < (KDIM / 2) * (MPAD * 2); i += BLOCK) (&Wp[0][0])[i] = 0.0f;
  __syncthreads();
  for (int t = tid; t < nnz; t += BLOCK) {
    const int k = mu1[t] * 16 + mu2[t];
    Wp[k >> 1][2 * mu3[t] + (k & 1)] = cg[t];   // unique (i1,i2,i3) triples
  }
  __syncthreads();

  const int iq = lane >> 2;     // staging: row within 8-row group
  const int q  = lane & 3;      // staging: 4-float chunk within row

  const int stride = GRID * WAVES;              // 8192 tiles per sweep
  int t = blockIdx.x * WAVES + wave;

  #pragma unroll 1
  for (int it = 0; it < COLTILES / stride; ++it, t += stride) {
    const int n  = t >> 3;                      // 128/16 = 8 column tiles per edge
    const int c0 = (t & 7) << 4;

    // ---- stage tiles: float4 global loads; y transposed for b64 pair reads ----
    const float* xbase = x + n * (DIM_IN * CHANNELS) + c0;
    const float* ybase = y + n * (DIM_IN * CHANNELS) + c0;
    #pragma unroll
    for (int i0 = 0; i0 < 2; ++i0) {
      const int i = i0 * 8 + iq;
      const float4 xv4 = *(const float4*)(xbase + i * CHANNELS + 4 * q);
      *(float4*)&xt[wave][i][4 * q] = xv4;      // 16B-aligned b128 store
      const float4 yv4 = *(const float4*)(ybase + i * CHANNELS + 4 * q);
      ytp[wave][4 * q + 0][i] = yv4.x;          // transposed scatter, conflict-free
      ytp[wave][4 * q + 1][i] = yv4.y;
      ytp[wave][4 * q + 2][i] = yv4.z;
      ytp[wave][4 * q + 3][i] = yv4.w;
    }
    __syncthreads();   // uniform: all waves run identical trip counts

    // ---- 112x16 = W(112x256) x z(256x16), exact f32 WMMA ----
    v8f acc[MT];
    #pragma unroll
    for (int m = 0; m < MT; ++m) acc[m] = (v8f){0.f, 0.f, 0.f, 0.f, 0.f, 0.f, 0.f, 0.f};

    #pragma unroll 2
    for (int kt = 0; kt < KT; ++kt) {
      // B fragment (4x16 f32): V0 = K = kb, V1 = K = kb+1; lane half gives kb offset 0/2
      const int kb = kt * 4 + (hi << 1);
      const int i1 = kb >> 4;          // same i1 for kb and kb+1 (kb&15 <= 14)
      const int i2 = kb & 15;          // even
      const float xv  = xt[wave][i1][j];
      const v2f   yv  = *(const v2f*)&ytp[wave][j][i2];     // b64 pair load
      const v2f   xvv = {xv, xv};
      const v2f   b   = xvv * yv;                           // v_pk_mul_f32

      #pragma unroll
      for (int m = 0; m < MT; ++m) {
        const v2f a = *(const v2f*)&Wp[kb >> 1][2 * (m * 16 + j)];  // b64 A pair
        acc[m] = __builtin_amdgcn_wmma_f32_16x16x4_f32(
            /*neg_a=*/false, a, /*neg_b=*/false, b,
            /*c_mod=*/(short)0, acc[m], /*reuse_a=*/false, /*reuse_b=*/false);
      }
    }

    // ---- store: VGPR r of tile m holds row m*16 + r + 8*hi, col c0+j ----
    float* op = out + n * (DIM_OUT * CHANNELS) + c0 + j;
    #pragma unroll
    for (int m = 0; m < MT; ++m) {
      #pragma unroll
      for (int r = 0; r < 8; ++r) {
        const int orow = m * 16 + (hi << 3) + r;
        if (orow < DIM_OUT) op[orow * CHANNELS] = acc[m][r];
      }
    }
  }
}

extern "C" void kernel_launch(void* const* d_in, const int* in_sizes, int n_in,
                              void* d_out, int out_size, void* d_ws, size_t ws_size,
                              hipStream_t stream) {
  const float* x   = (const float*)d_in[0];
  const float* y   = (const float*)d_in[1];
  const int*   mu1 = (const int*)d_in[2];
  const int*   mu2 = (const int*)d_in[3];
  const int*   mu3 = (const int*)d_in[4];
  const float* cg  = (const float*)d_in[5];
  const int    nnz = in_sizes[5];

  tp_cg_wmma_kernel<<<GRID, BLOCK, 0, stream>>>(x, y, mu1, mu2, mu3, cg, nnz,
                                                (float*)d_out);
}